// RepComposer_13365938225808
// MI455X (gfx1250) — compile-verified
//
#include <hip/hip_runtime.h>

typedef __attribute__((ext_vector_type(2))) float v2f;
typedef __attribute__((ext_vector_type(8))) float v8f;

#define DD   128
#define ROWS 16
#define WS   136   // padded row stride (floats) for weight tiles in LDS
#define ZS   132   // padded row stride (floats) for z tile in LDS

// ---- gfx1250 async VMEM->LDS (ASYNCcnt-tracked), GVS addressing ----------
__device__ __forceinline__ void async_ld_b128(unsigned lds_addr, unsigned goff,
                                              const void* base) {
    asm volatile("global_load_async_to_lds_b128 %0, %1, %2"
                 :: "v"(lds_addr), "v"(goff), "s"(base)
                 : "memory");
}

__device__ __forceinline__ void wait_async0() {
#if __has_builtin(__builtin_amdgcn_s_wait_asynccnt)
    __builtin_amdgcn_s_wait_asynccnt(0);
#else
    asm volatile("s_wait_asynccnt 0" ::: "memory");
#endif
}

// -------------------------------------------------------------------------
// Scatter-add: agg[dst[e], :] += x[src[e], :]   (32 lanes x float4 per edge)
// -------------------------------------------------------------------------
__global__ __launch_bounds__(256)
void gin_scatter(const float* __restrict__ x, const int* __restrict__ src,
                 const int* __restrict__ dst, float* __restrict__ agg, int E) {
    long long tid = (long long)blockIdx.x * blockDim.x + threadIdx.x;
    int e = (int)(tid >> 5);
    if (e >= E) return;
    int q = ((int)tid & 31) << 2;          // feature offset (0..124 step 4)
    int s = src[e];
    int d = dst[e];
    float4 v = *(const float4*)(x + (size_t)s * DD + q);
    float* o = agg + (size_t)d * DD + q;
    unsafeAtomicAdd(o + 0, v.x);
    unsafeAtomicAdd(o + 1, v.y);
    unsafeAtomicAdd(o + 2, v.z);
    unsafeAtomicAdd(o + 3, v.w);
}

// -------------------------------------------------------------------------
// Fused MLP: z = x + agg ; h1 = relu(z@W1+b1) ; z2 = h1@W2+b2
// 256 threads = 8 wave32, 16 rows per block, wave w owns cols [16w,16w+16).
// f32 WMMA 16x16x4, K=128 -> 32 wmma per GEMM per wave.
// Weights async-staged into LDS (global_load_async_to_lds_b128).
// Also accumulates per-column sum / sumsq for BatchNorm.
// -------------------------------------------------------------------------
__global__ __launch_bounds__(256)
void gin_mlp(const float* __restrict__ x, const float* __restrict__ agg,
             const float* __restrict__ W1, const float* __restrict__ b1,
             const float* __restrict__ W2, const float* __restrict__ b2,
             float* __restrict__ z2out, float* __restrict__ sums,
             float* __restrict__ sumsq) {
    extern __shared__ float smem[];
    float* w1s = smem;                 // WS*128 floats (padded rows)
    float* w2s = smem + WS * DD;       // WS*128 floats
    float* zt  = smem + 2 * WS * DD;   // ZS*16 floats

    const int tid  = threadIdx.x;
    const int row0 = blockIdx.x * ROWS;

    // Kick off async weight staging: 16B per lane per op, ASYNCcnt-tracked.
    {
        const unsigned w1b = (unsigned)(size_t)(void*)w1s;
        const unsigned w2b = (unsigned)(size_t)(void*)w2s;
        for (int i = tid; i < (DD * DD) / 4; i += 256) {
            int k = i >> 5, c = i & 31;                       // row k, float4 col c
            unsigned ldsoff = (unsigned)((k * WS + c * 4) * 4);
            unsigned goff   = (unsigned)(i * 16);
            async_ld_b128(w1b + ldsoff, goff, W1);
            async_ld_b128(w2b + ldsoff, goff, W2);
        }
    }

    // Stage z tile = x + agg (overlaps with async weight DMA), padded rows
    {
        const float4* xg = (const float4*)(x   + (size_t)row0 * DD);
        const float4* ag = (const float4*)(agg + (size_t)row0 * DD);
        float4* zl = (float4*)zt;
        for (int i = tid; i < ROWS * 32; i += 256) {
            int r = i >> 5, c = i & 31;
            float4 a = xg[i];
            float4 b = ag[i];
            float4 v;
            v.x = a.x + b.x; v.y = a.y + b.y; v.z = a.z + b.z; v.w = a.w + b.w;
            zl[r * (ZS / 4) + c] = v;
        }
    }
    wait_async0();
    __syncthreads();

    const int lane  = tid & 31;
    const int wv    = tid >> 5;
    const int n     = lane & 15;              // A's M index == B's N index
    const int kk    = (lane >> 4) << 1;       // K sub-pair: lanes 0-15 -> {0,1}, 16-31 -> {2,3}
    const int col   = (wv << 4) + n;          // global output column
    const int mbase = (lane >> 4) << 3;       // C rows: 0..7 or 8..15

    const float bias1 = b1[col];
    const float bias2 = b2[col];

    // ---- GEMM1: h1 = relu(z @ W1 + b1) ----
    v8f c1 = {};
    for (int k0 = 0; k0 < DD; k0 += 4) {
        v2f a, b;
        a.x = zt[n * ZS + k0 + kk];
        a.y = zt[n * ZS + k0 + kk + 1];
        b.x = w1s[(k0 + kk) * WS + col];
        b.y = w1s[(k0 + kk + 1) * WS + col];
        c1 = __builtin_amdgcn_wmma_f32_16x16x4_f32(
                 false, a, false, b, (short)0, c1, false, false);
    }
    __syncthreads();   // all waves done reading z tile
    for (int r = 0; r < 8; ++r) {
        float v = c1[r] + bias1;
        zt[(mbase + r) * ZS + col] = v > 0.0f ? v : 0.0f;
    }
    __syncthreads();   // h1 tile complete in LDS

    // ---- GEMM2: z2 = h1 @ W2 + b2 ----
    v8f c2 = {};
    for (int k0 = 0; k0 < DD; k0 += 4) {
        v2f a, b;
        a.x = zt[n * ZS + k0 + kk];
        a.y = zt[n * ZS + k0 + kk + 1];
        b.x = w2s[(k0 + kk) * WS + col];
        b.y = w2s[(k0 + kk + 1) * WS + col];
        c2 = __builtin_amdgcn_wmma_f32_16x16x4_f32(
                 false, a, false, b, (short)0, c2, false, false);
    }
    __syncthreads();   // all waves done reading h1 tile before overwrite

    float* zo = z2out + (size_t)row0 * DD;
    for (int r = 0; r < 8; ++r) {
        float v = c2[r] + bias2;
        zt[(mbase + r) * ZS + col] = v;   // keep for stats
        zo[(mbase + r) * DD + col] = v;   // pre-BN activations to global
    }
    __syncthreads();

    // ---- BatchNorm partials: per-column sum (threads 0-127) / sumsq (128-255)
    if (tid < DD) {
        float s = 0.0f;
        for (int r = 0; r < ROWS; ++r) s += zt[r * ZS + tid];
        unsafeAtomicAdd(&sums[tid], s);
    } else {
        int cc = tid - DD;
        float q = 0.0f;
        for (int r = 0; r < ROWS; ++r) { float v = zt[r * ZS + cc]; q += v * v; }
        unsafeAtomicAdd(&sumsq[cc], q);
    }
}

// -------------------------------------------------------------------------
// Fold stats into scale/bias:  scale = gamma*rsqrt(var+eps), bias = beta-mean*scale
// -------------------------------------------------------------------------
__global__ void gin_bnstats(const float* __restrict__ sums,
                            const float* __restrict__ sumsq,
                            const float* __restrict__ gamma,
                            const float* __restrict__ beta,
                            float* __restrict__ sb, int N) {
    int c = threadIdx.x;
    float invN = 1.0f / (float)N;
    float mean = sums[c] * invN;
    float var  = sumsq[c] * invN - mean * mean;   // biased, matches jnp.var
    float inv  = rsqrtf(var + 1e-5f);
    float sc   = gamma[c] * inv;
    sb[c]       = sc;
    sb[c + DD]  = beta[c] - mean * sc;
}

// -------------------------------------------------------------------------
// In-place BN + ReLU on the output slice (float4)
// -------------------------------------------------------------------------
__global__ __launch_bounds__(256)
void gin_bnapply(float* __restrict__ z, const float* __restrict__ sb, int n4) {
    int i = blockIdx.x * 256 + threadIdx.x;
    if (i >= n4) return;
    int c4 = (i & 31) << 2;
    float4 v = ((const float4*)z)[i];
    float4 r;
    r.x = fmaxf(v.x * sb[c4 + 0] + sb[DD + c4 + 0], 0.0f);
    r.y = fmaxf(v.y * sb[c4 + 1] + sb[DD + c4 + 1], 0.0f);
    r.z = fmaxf(v.z * sb[c4 + 2] + sb[DD + c4 + 2], 0.0f);
    r.w = fmaxf(v.w * sb[c4 + 3] + sb[DD + c4 + 3], 0.0f);
    ((float4*)z)[i] = r;
}

// -------------------------------------------------------------------------
extern "C" void kernel_launch(void* const* d_in, const int* in_sizes, int n_in,
                              void* d_out, int out_size, void* d_ws, size_t ws_size,
                              hipStream_t stream) {
    const float* h     = (const float*)d_in[0];
    const int*   ei    = (const int*)d_in[1];
    const float* W1    = (const float*)d_in[2];
    const float* b1    = (const float*)d_in[3];
    const float* W2    = (const float*)d_in[4];
    const float* b2    = (const float*)d_in[5];
    const float* gamma = (const float*)d_in[6];
    const float* beta  = (const float*)d_in[7];

    const int N = in_sizes[0] / DD;     // 50000
    const int E = in_sizes[1] / 2;      // 800000
    const int* src = ei;
    const int* dst = ei + E;

    float* out   = (float*)d_out;                    // [2, N, 128]
    float* agg   = (float*)d_ws;                     // [N, 128]
    float* sums  = agg + (size_t)N * DD;             // [128]
    float* sumsq = sums + DD;                        // [128]
    float* sb    = sumsq + DD;                       // [256] scale|bias

    const size_t wstep = (size_t)DD * DD;
    const size_t sliceBytes = (size_t)N * DD * sizeof(float);
    const int scatterBlocks = (int)(((long long)E * 32 + 255) / 256);
    const int applyBlocks   = (N * (DD / 4) + 255) / 256;
    const size_t ldsBytes   = (2 * WS * DD + ZS * ROWS) * sizeof(float); // ~144 KB

    for (int l = 0; l < 2; ++l) {
        const float* xin = (l == 0) ? h : out;        // layer 2 reads layer-1 output slice
        float* z2 = out + (size_t)l * N * DD;

        hipMemsetAsync(agg, 0, sliceBytes, stream);
        hipMemsetAsync(sums, 0, 2 * DD * sizeof(float), stream);

        gin_scatter<<<scatterBlocks, 256, 0, stream>>>(xin, src, dst, agg, E);
        gin_mlp<<<N / ROWS, 256, ldsBytes, stream>>>(
            xin, agg, W1 + l * wstep, b1 + l * DD, W2 + l * wstep, b2 + l * DD,
            z2, sums, sumsq);
        gin_bnstats<<<1, DD, 0, stream>>>(sums, sumsq, gamma + l * DD, beta + l * DD, sb, N);
        gin_bnapply<<<applyBlocks, 256, 0, stream>>>(z2, sb, N * (DD / 4));
    }
}